// FFTDepthwiseConv3D_19628000543406
// MI455X (gfx1250) — compile-verified
//
#include <hip/hip_runtime.h>

// ---- problem constants (B=2,T=8 -> 16 independent frames; KT=1 => T is identity) ----
#define FF   16      // B*T frames
#define HH   224
#define WW   224
#define CC   96
#define KK   7
#define RR   3       // radius
#define TILE 16      // output tile (h and w)
#define HDIM 22      // TILE + 2*RR (halo tile edge)
#define CPW  16      // channels per workgroup
#define CSTR 17      // padded LDS channel stride (16 + 1 dword TDM pad)
#define NPOS (HDIM * HDIM)        // 484 halo positions
#define XS_FLOATS (NPOS * CSTR)   // 8228
#define OS_FLOATS (TILE * TILE * CSTR) // 4352
#define WK_FLOATS (CPW * 52)      // 832 (49 flipped taps + zero slot + pad)
#define NFRAG (KK * 6)            // 42 WMMA fragments per channel

typedef __attribute__((ext_vector_type(2))) float     v2f;
typedef __attribute__((ext_vector_type(8))) float     v8f;
typedef __attribute__((ext_vector_type(4))) unsigned  v4u;
typedef __attribute__((ext_vector_type(8))) int       v8i;
typedef __attribute__((ext_vector_type(4))) int       v4i;

#if defined(__HIP_DEVICE_COMPILE__) && defined(__gfx1250__) && \
    __has_builtin(__builtin_amdgcn_tensor_load_to_lds) &&      \
    __has_builtin(__builtin_amdgcn_s_wait_tensorcnt)
#define HAVE_TDM 1
#else
#define HAVE_TDM 0
#endif

// compile-time probe so stderr tells us which load path the device pass took
#if defined(__HIP_DEVICE_COMPILE__) && defined(__gfx1250__)
#if HAVE_TDM
#pragma message("CDNA5 probe: tensor_load_to_lds AVAILABLE -> TDM path compiled in")
#else
#warning "CDNA5 probe: tensor_load_to_lds NOT available -> manual load path only"
#endif
#endif

__global__ __launch_bounds__(128) void fftdw_conv7x7_kernel(
    const float* __restrict__ x,      // (F, H, W, C) fp32
    const float* __restrict__ kern,   // (C, 1, 7, 7) fp32
    float* __restrict__ out)          // (F, H, W, C) fp32
{
  extern __shared__ float smem[];
  float* xs = smem;                         // halo tile: [pos(22x22)][CSTR]
  float* os = smem + XS_FLOATS;             // out tile : [pos(16x16)][CSTR]
  float* wk = smem + XS_FLOATS + OS_FLOATS; // weights  : [CPW][52] (idx49 == 0)

  const int tid    = threadIdx.x;
  const int lane   = tid & 31;
  const int wave   = tid >> 5;
  const int lane16 = tid & 15;
  const int tx = blockIdx.x % 14, ty = blockIdx.x / 14;
  const int f  = blockIdx.y;
  const int cg0 = blockIdx.z * CPW;
  const int h0 = ty * TILE, w0 = tx * TILE;

  __builtin_prefetch(&kern[(long)cg0 * 49], 0, 0); // global_prefetch_b8 on weights

  const bool interior = (tx >= 1) && (tx <= 12) && (ty >= 1) && (ty <= 12);
#if HAVE_TDM
  const bool use_tdm = interior; // TDM OOB = zero-fill, cannot express circular wrap
#else
  const bool use_tdm = false;
#endif

#if HAVE_TDM
  // ---- issue the async tile DMA FIRST so it overlaps the weight staging ----
  if (use_tdm && wave == 0) { // one TDM op per workgroup; wave-level, uniform branch
    unsigned long long gaddr = (unsigned long long)(const void*)x +
        (unsigned long long)(((((long)f * HH + (h0 - RR)) * WW + (w0 - RR)) * CC + cg0) * 4l);
    unsigned ldsb = (unsigned)__builtin_amdgcn_groupstaticsize(); // dyn-LDS base
    // D# group0: count=1 | lds_addr | global_addr[56:0] | type=2
    v4u g0 = { 1u,
               ldsb,
               (unsigned)(gaddr & 0xffffffffull),
               (unsigned)((gaddr >> 32) & 0x01ffffffull) | (2u << 30) };
    // D# group1: data_size=4B, pad_enable, pad_interval=16dw, pad_amount=1dw
    //            tensor: dim0=96(C), dim1=224(W); tile: 16ch x 22w x 22h
    //            dim0_stride=96 (next w pos), dim1_stride=21504 (next h row)
    v8i g1 = { (int)((2u << 16) | (1u << 20) | (3u << 22)),
               (int)(96u << 16),
               (int)(224u << 16),
               (int)(16u << 16),
               (int)(22u | (22u << 16)),
               (int)96,
               (int)(21504u << 16),
               0 };
    v4i g2 = { 224, 0, 0, 0 }; // tensor_dim2 = H; tile_dim3 = 0
    v4i g3 = { 0, 0, 0, 0 };
#if defined(__clang_major__) && (__clang_major__ >= 23)
    v8i g4 = { 0, 0, 0, 0, 0, 0, 0, 0 };
    __builtin_amdgcn_tensor_load_to_lds(g0, g1, g2, g3, g4, 0);
#else
    __builtin_amdgcn_tensor_load_to_lds(g0, g1, g2, g3, 0);
#endif
  }
#endif

  // ---- stage flipped 7x7 weights into LDS (freq-multiply == circular conv with
  //      center-aligned flipped kernel: out(h,w) = sum k(6-i,6-j) x(h+i-3, w+j-3)) ----
  for (int idx = tid; idx < CPW * 49; idx += 128) {
    int cl = idx / 49, r = idx - cl * 49;
    wk[cl * 52 + r] = kern[(long)(cg0 + cl) * 49 + (48 - r)];
  }
  if (tid < CPW) wk[tid * 52 + 49] = 0.f; // zero slot for out-of-band Toeplitz entries

  // ---- finish staging the 22x22x16ch halo tile ----
  if (use_tdm) {
#if HAVE_TDM
    if (wave == 0) __builtin_amdgcn_s_wait_tensorcnt(0);
#endif
  } else {
    // boundary tiles: coalesced manual load with circular wrap (64B per half-wave)
    for (int p = tid >> 4; p < NPOS; p += 8) {
      int hl = p / HDIM, wl = p - hl * HDIM;
      int gh = h0 + hl - RR; gh += (gh < 0) ? HH : 0; gh -= (gh >= HH) ? HH : 0;
      int gw = w0 + wl - RR; gw += (gw < 0) ? WW : 0; gw -= (gw >= WW) ? WW : 0;
      xs[p * CSTR + lane16] =
          x[(((long)f * HH + gh) * WW + gw) * CC + cg0 + lane16];
    }
  }
  __syncthreads();

  // ---- compute: per channel, D(16x16) = sum_i  X_i(16x22) * Toeplitz_i(22x16).
  //      Two channels per wave in flight: two independent WMMA accumulator chains
  //      so every s_wait_dscnt slack slot is filled by the sibling chain's WMMA. ----
  const int half = lane >> 4;       // A/B K-halves per ISA 32-bit fragment layout
  const int mn   = lane & 15;       // A row (M) / B col (N)
  const int qh   = half * 2;

  for (int sp = 0; sp < 2; ++sp) {
    const int clA = wave * 4 + sp * 2; // channel pair owned by this wave
    const int clB = clA + 1;
    const float* wkA = &wk[clA * 52];
    const float* wkB = &wk[clB * 52];

    v8f accA = { 0.f, 0.f, 0.f, 0.f, 0.f, 0.f, 0.f, 0.f };
    v8f accB = { 0.f, 0.f, 0.f, 0.f, 0.f, 0.f, 0.f, 0.f };

#pragma unroll
    for (int t = 0; t < NFRAG; ++t) {
      const int i = t / 6, kc = t - 6 * i;           // kernel row / K chunk
      const int q0 = kc * 4 + qh;                    // K index (w' within 0..23)
      const int qa = (q0     <= 21) ? q0     : 21;   // clamp pad cols (B==0 there)
      const int qb = (q0 + 1 <= 21) ? q0 + 1 : 21;
      const int rowb = (mn + i) * HDIM;
      const int pa = (rowb + qa) * CSTR;
      const int pb = (rowb + qb) * CSTR;
      const int jx = q0 - mn, jy = jx + 1;           // tap index = q - n
      const int ixw = ((unsigned)jx < 7u) ? (i * KK + jx) : 49;
      const int iyw = ((unsigned)jy < 7u) ? (i * KK + jy) : 49;

      v2f aA, aB, bA, bB;
      aA.x = xs[pa + clA];  aA.y = xs[pb + clA];
      aB.x = xs[pa + clB];  aB.y = xs[pb + clB];
      bA.x = wkA[ixw];      bA.y = wkA[iyw];
      bB.x = wkB[ixw];      bB.y = wkB[iyw];

      accA = __builtin_amdgcn_wmma_f32_16x16x4_f32(
          false, aA, false, bA, (short)0, accA, false, false);
      accB = __builtin_amdgcn_wmma_f32_16x16x4_f32(
          false, aB, false, bB, (short)0, accB, false, false);
    }

#pragma unroll
    for (int r = 0; r < 8; ++r) {                    // D: VGPR r -> M = r + 8*half, N = mn
      const int M = r + half * 8;
      os[(M * TILE + mn) * CSTR + clA] = accA[r];
      os[(M * TILE + mn) * CSTR + clB] = accB[r];
    }
  }
  __syncthreads();

  // ---- coalesced store: 16 contiguous channels (64B) per half-wave ----
  for (int p = tid >> 4; p < TILE * TILE; p += 8) {
    const int M = p >> 4, N = p & 15;
    out[(((long)f * HH + (h0 + M)) * WW + (w0 + N)) * CC + cg0 + lane16] =
        os[p * CSTR + lane16];
  }
}

extern "C" void kernel_launch(void* const* d_in, const int* in_sizes, int n_in,
                              void* d_out, int out_size, void* d_ws, size_t ws_size,
                              hipStream_t stream) {
  const float* x    = (const float*)d_in[0]; // (2,8,224,224,96) fp32
  const float* kern = (const float*)d_in[1]; // (96,1,7,7) fp32
  float* out        = (float*)d_out;         // (2,8,224,224,96) fp32
  (void)in_sizes; (void)n_in; (void)out_size; (void)d_ws; (void)ws_size;

  dim3 grid(14 * 14, FF, CC / CPW);  // 196 tiles x 16 frames x 6 channel groups
  dim3 block(128);                   // 4 waves (wave32)
  size_t shmem = (size_t)(XS_FLOATS + OS_FLOATS + WK_FLOATS) * sizeof(float); // ~53.6 KB
  fftdw_conv7x7_kernel<<<grid, block, shmem, stream>>>(x, kern, out);
}